// S4D_46007689675190
// MI455X (gfx1250) — compile-verified
//
#include <hip/hip_runtime.h>

typedef float v2f __attribute__((ext_vector_type(2)));
typedef float v8f __attribute__((ext_vector_type(8)));

namespace {
constexpr int kH = 512;
constexpr int kL = 4096;
constexpr int kN = 32;               // states per head (N/2)
constexpr int kT = 16;               // chunk length  (WMMA tile rows)
constexpr int kS = 16;               // super-chunks  (WMMA tile cols)
constexpr int kSupLen = kL / kS;     // 256
constexpr int kNCh = kSupLen / kT;   // 16 chunks per super-chunk

constexpr int UPAD = 20;             // U_l[col][j]
constexpr int SPAD = 34;             // S_l[col][n], Sin_l[col][n]
constexpr int PPAD = 34;             // P_l[i][n]
constexpr int WPAD = 18;             // W_l[n][j]
constexpr int YSTR = 260;            // y_l stride per super-chunk (pad vs 256)
}

__device__ __forceinline__ v8f wmma4(v2f a, v2f b, v8f c) {
  // D(16x16,f32) = A(16x4,f32) x B(4x16,f32) + C
  return __builtin_amdgcn_wmma_f32_16x16x4_f32(false, a, false, b,
                                               (short)0, c, false, false);
}

__global__ __launch_bounds__(32, 1) void s4d_scan_kernel(
    const float* __restrict__ u, const float* __restrict__ Cmat,
    const float* __restrict__ log_dt, const float* __restrict__ log_A_real,
    float* __restrict__ y)
{
  __shared__ float U_l[kS * UPAD];
  __shared__ float S_l[kS * SPAD];
  __shared__ float Sin_l[kS * SPAD];
  __shared__ float P_l[kT * PPAD];
  __shared__ float W_l[kN * WPAD];
  __shared__ float Kloc[kT];
  __shared__ float rT16_l[kN];
  __shared__ float rT256_l[kN];
  __shared__ float y_l[kS * YSTR];

  const int h = blockIdx.x;
  const int b = blockIdx.y;
  const int lane = threadIdx.x;        // single wave32 workgroup

  // ---------- per-head constants: lane owns state n = lane ----------
  {
    const int n = lane;
    const float dt = __expf(log_dt[h]);
    const float A  = -__expf(log_A_real[h * kN + n]);
    const float r  = __expf(A * dt);
    const float Cr = Cmat[(h * kN + n) * 2] * (r - 1.0f) / A;
    float rT16 = r * r; rT16 *= rT16; rT16 *= rT16; rT16 *= rT16;                 // r^16
    float rT256 = rT16 * rT16; rT256 *= rT256; rT256 *= rT256; rT256 *= rT256;    // r^256
    rT16_l[n]  = rT16;
    rT256_l[n] = rT256;
    float p = 1.0f;
    for (int j = kT - 1; j >= 0; --j) { W_l[n * WPAD + j] = p; p *= r; }          // r^(15-j)
    p = r;
    for (int i = 0; i < kT; ++i) { P_l[i * PPAD + n] = Cr * p; p *= r; }          // Cr*r^(i+1)
    Sin_l[n] = Cr;                     // stash Cr for Kloc[0]
  }
  __syncthreads();
  if (lane < kT) {                     // Kloc[d] = sum_n Cr_n * r_n^d
    float s = 0.f;
    if (lane == 0) { for (int n = 0; n < kN; ++n) s += Sin_l[n]; }
    else           { for (int n = 0; n < kN; ++n) s += P_l[(lane - 1) * PPAD + n]; }
    Kloc[lane] = s;
  }
  for (int k = lane; k < kS * SPAD; k += 32) S_l[k] = 0.f;
  __syncthreads();

  const float* ug = u + ((size_t)b * kH + h) * kL;
  float*       yg = y + ((size_t)b * kH + h) * kL;

  const int colb = lane & 15;          // row index i for A, column index for B/D
  const int ioff = (lane >> 4) * 8;    // D-tile row offset (lanes 16..31 -> M+8)
  const int kofs = (lane >> 4) * 2;    // K offset for A/B operands (lanes 16..31 -> K+2)

  // ---------- main loop: 16 chunks, 16 super-chunk columns in parallel ----------
  for (int c = 0; c < kNCh; ++c) {
    {   // load U tile: element (col=s, j) = u[s*256 + c*16 + j]
      const int s = lane >> 1, half = lane & 1;
      const float* src = ug + s * kSupLen + c * kT + half * 8;
      float4 a0 = *(const float4*)(src);
      float4 a1 = *(const float4*)(src + 4);
      float* dst = &U_l[s * UPAD + half * 8];
      *(float4*)(dst)     = a0;
      *(float4*)(dst + 4) = a1;
      __builtin_prefetch(src + kT, 0, 0);          // next chunk's tile
    }
    __syncthreads();

    v8f acc = {0.f, 0.f, 0.f, 0.f, 0.f, 0.f, 0.f, 0.f};
    // carry: y += P(16i x 32n) @ S(32n x 16col)
#pragma unroll
    for (int nb = 0; nb < 8; ++nb) {
      const int n0 = nb * 4 + kofs;
      v2f a;  a.x  = P_l[colb * PPAD + n0];  a.y  = P_l[colb * PPAD + n0 + 1];
      v2f bb; bb.x = S_l[colb * SPAD + n0];  bb.y = S_l[colb * SPAD + n0 + 1];
      acc = wmma4(a, bb, acc);
    }
    // intra-chunk: y += LowerToeplitz(Kloc)(16i x 16j) @ U(16j x 16col)
    v2f bu[4];
#pragma unroll
    for (int jb = 0; jb < 4; ++jb) {
      const int j0 = jb * 4 + kofs;
      v2f a;
      a.x = (colb >= j0)     ? Kloc[colb - j0]     : 0.f;
      a.y = (colb >= j0 + 1) ? Kloc[colb - j0 - 1] : 0.f;
      v2f bb; bb.x = U_l[colb * UPAD + j0]; bb.y = U_l[colb * UPAD + j0 + 1];
      bu[jb] = bb;
      acc = wmma4(a, bb, acc);
    }
#pragma unroll
    for (int v = 0; v < 8; ++v) y_l[colb * YSTR + c * kT + ioff + v] = acc[v];

    // state: S = diag(r^16)*S + W(32n x 16j) @ U(16j x 16col)   (two 16-row tiles)
    v8f st0 = {0.f, 0.f, 0.f, 0.f, 0.f, 0.f, 0.f, 0.f};
    v8f st1 = {0.f, 0.f, 0.f, 0.f, 0.f, 0.f, 0.f, 0.f};
#pragma unroll
    for (int jb = 0; jb < 4; ++jb) {
      const int j0 = jb * 4 + kofs;
      v2f a0; a0.x = W_l[colb * WPAD + j0];        a0.y = W_l[colb * WPAD + j0 + 1];
      v2f a1; a1.x = W_l[(16 + colb) * WPAD + j0]; a1.y = W_l[(16 + colb) * WPAD + j0 + 1];
      st0 = wmma4(a0, bu[jb], st0);
      st1 = wmma4(a1, bu[jb], st1);
    }
    __syncthreads();
#pragma unroll
    for (int v = 0; v < 8; ++v) {
      const int n0 = ioff + v, n1 = n0 + 16;
      S_l[colb * SPAD + n0] = rT16_l[n0] * S_l[colb * SPAD + n0] + st0[v];
      S_l[colb * SPAD + n1] = rT16_l[n1] * S_l[colb * SPAD + n1] + st1[v];
    }
    __syncthreads();
  }

  // ---------- combine super-chunks: Sin[s] = r^256 * Sin[s-1] + Send[s-1] ----------
  {
    const int n = lane;
    float sin = 0.f;
    for (int s = 0; s < kS; ++s) {
      Sin_l[s * SPAD + n] = sin;
      sin = rT256_l[n] * sin + S_l[s * SPAD + n];
    }
  }
  __syncthreads();

  // ---------- fix-up: y[s*256 + c*16 + i] += sum_n C_n r^{c*16+i+1} Sin[s][n] ----------
  for (int c = 0; c < kNCh; ++c) {
    v8f acc;
#pragma unroll
    for (int v = 0; v < 8; ++v) acc[v] = y_l[colb * YSTR + c * kT + ioff + v];
#pragma unroll
    for (int nb = 0; nb < 8; ++nb) {
      const int n0 = nb * 4 + kofs;
      v2f a;  a.x  = P_l[colb * PPAD + n0];    a.y  = P_l[colb * PPAD + n0 + 1];
      v2f bb; bb.x = Sin_l[colb * SPAD + n0];  bb.y = Sin_l[colb * SPAD + n0 + 1];
      acc = wmma4(a, bb, acc);
    }
#pragma unroll
    for (int v = 0; v < 8; ++v) y_l[colb * YSTR + c * kT + ioff + v] = acc[v];
    __syncthreads();
    for (int i = 0; i < kT; ++i) P_l[i * PPAD + lane] *= rT16_l[lane];   // advance to next chunk's powers
    __syncthreads();
  }

  // ---------- coalesced store ----------
  for (int it = 0; it < kL / 128; ++it) {
    const int idx = it * 128 + lane * 4;
    const int s = idx >> 8, rrem = idx & 255;
    float4 val = *(const float4*)&y_l[s * YSTR + rrem];
    *(float4*)(yg + idx) = val;
  }
}

extern "C" void kernel_launch(void* const* d_in, const int* in_sizes, int n_in,
                              void* d_out, int out_size, void* d_ws, size_t ws_size,
                              hipStream_t stream) {
  (void)n_in; (void)out_size; (void)d_ws; (void)ws_size;
  const float* u       = (const float*)d_in[0];
  const float* C       = (const float*)d_in[1];
  const float* log_dt  = (const float*)d_in[2];
  const float* log_A   = (const float*)d_in[3];
  float* y = (float*)d_out;
  const int B = in_sizes[0] / (kH * kL);   // 16
  dim3 grid(kH, B);
  s4d_scan_kernel<<<grid, 32, 0, stream>>>(u, C, log_dt, log_A, y);
}